// PatchGuardPooling_33964601376868
// MI455X (gfx1250) — compile-verified
//
#include <hip/hip_runtime.h>

// ---------------- CDNA5 (gfx1250) PatchGuardPooling ----------------
// x: [64,1000,24,24] f32. Per channel: relu, total sum, max 6x6 box sum,
// logit = total - maxbox; then argmax over 1000 classes per batch.

typedef __attribute__((ext_vector_type(2))) float v2f;
typedef __attribute__((ext_vector_type(8))) float v8f;
typedef __attribute__((ext_vector_type(4))) int   v4i;

#define GLOBAL_AS __attribute__((address_space(1)))
#define SHARED_AS __attribute__((address_space(3)))

#define NB 64
#define NC 1000
#define HH 24
#define WW 24
#define KS 6
#define OH 19
#define OW 19
#define TILE (HH * WW)          // 576 floats = 2304 bytes = 144 x 16B
#define WAVES_PER_BLOCK 8
#define NBC (NB * NC)           // 64000 channels

__global__ __launch_bounds__(256) void pgp_channel_kernel(
    const float* __restrict__ x, float* __restrict__ logits) {
  __shared__ float lds_tile[WAVES_PER_BLOCK][TILE];     // relu'd 24x24 tile
  __shared__ float lds_v[WAVES_PER_BLOCK][OH * WW];     // vertical box sums 19x24

  const int lane = threadIdx.x & 31;
  const int wave = threadIdx.x >> 5;
  const int bc = blockIdx.x * WAVES_PER_BLOCK + wave;   // 0..63999

  const float* gsrc = x + (size_t)bc * TILE;
  float* tile = &lds_tile[wave][0];

  // ---- Stage 2304B tile: async global->LDS, 144 x B128 transfers ----
  {
    GLOBAL_AS v4i* g = (GLOBAL_AS v4i*)(GLOBAL_AS char*)(char*)(const_cast<float*>(gsrc));
    SHARED_AS v4i* l = (SHARED_AS v4i*)(SHARED_AS char*)(char*)tile;
#pragma unroll
    for (int j = 0; j < 5; ++j) {
      int idx = j * 32 + lane;            // 0..159, valid < 144
      if (idx < 144) {
        __builtin_amdgcn_global_load_async_to_lds_b128(g + idx, l + idx, 0, 0);
      }
    }
  }
  __builtin_amdgcn_s_wait_asynccnt(0);
  asm volatile("" ::: "memory");

  // ---- relu in place + per-channel total sum (wave32 reduce) ----
  float tot = 0.f;
#pragma unroll
  for (int j = 0; j < TILE / 32; ++j) {   // 18 per lane
    int idx = j * 32 + lane;
    float v = tile[idx];
    v = fmaxf(v, 0.f);
    tile[idx] = v;
    tot += v;
  }
#pragma unroll
  for (int off = 16; off > 0; off >>= 1) tot += __shfl_xor(tot, off, 32);

  // ---- Vertical 6-row box sums via WMMA: V(19x24) = S^T(19x24) x Xr(24x24) ----
  // A (16x4 f32): lane l -> M = l%16, K = v + 2*(l/16); band: 1 iff m <= k < m+6
  // B (4x16 f32): lane l -> N = l%16, K = v + 2*(l/16)
  float* vp = &lds_v[wave][0];
  const int lm = lane & 15;
  const int lh = lane >> 4;
#pragma unroll
  for (int mt = 0; mt < 2; ++mt) {
#pragma unroll
    for (int nt = 0; nt < 2; ++nt) {
      v8f acc = {};
#pragma unroll
      for (int ks = 0; ks < 6; ++ks) {
        v2f a, b;
#pragma unroll
        for (int v = 0; v < 2; ++v) {
          int kg = ks * 4 + v + 2 * lh;                 // input row 0..23
          int m = mt * 16 + lm;                         // output row
          a[v] = (m < OH && kg >= m && kg < m + KS) ? 1.0f : 0.0f;
          int n = nt * 16 + lm;                         // column
          int nn = (n < WW) ? n : 0;
          float bv = tile[kg * WW + nn];
          b[v] = (n < WW) ? bv : 0.0f;
        }
        acc = __builtin_amdgcn_wmma_f32_16x16x4_f32(
            false, a, false, b, (short)0, acc, false, false);
      }
      // D layout: VGPR i, lanes 0-15 -> M = i; lanes 16-31 -> M = 8+i; N = l%16
#pragma unroll
      for (int i = 0; i < 8; ++i) {
        int m = mt * 16 + i + (lh << 3);
        int n = nt * 16 + lm;
        if (m < OH && n < WW) vp[m * WW + n] = acc[i];
      }
    }
  }

  // ---- Horizontal 6-wide sums + max over 19x19 windows ----
  float lmax = -3.402823466e38f;
  for (int o = lane; o < OH * OW; o += 32) {
    int r = o / OW;
    int c = o - r * OW;
    float s = 0.f;
#pragma unroll
    for (int j = 0; j < KS; ++j) s += vp[r * WW + c + j];
    lmax = fmaxf(lmax, s);
  }
#pragma unroll
  for (int off = 16; off > 0; off >>= 1)
    lmax = fmaxf(lmax, __shfl_xor(lmax, off, 32));

  if (lane == 0) logits[bc] = tot - lmax;
}

// ---- argmax over classes, first-occurrence tie-break (jnp.argmax) ----
__global__ __launch_bounds__(256) void pgp_argmax_kernel(
    const float* __restrict__ logits, float* __restrict__ out) {
  __shared__ float svals[256];
  __shared__ int sidx[256];
  const int b = blockIdx.x;
  const int t = threadIdx.x;

  float best = -3.402823466e38f;
  int bidx = 0;
  for (int c = t; c < NC; c += 256) {
    float v = logits[b * NC + c];
    if (v > best) { best = v; bidx = c; }
  }
  svals[t] = best;
  sidx[t] = bidx;
  __syncthreads();
#pragma unroll
  for (int s = 128; s > 0; s >>= 1) {
    if (t < s) {
      float v2 = svals[t + s];
      int i2 = sidx[t + s];
      if (v2 > svals[t] || (v2 == svals[t] && i2 < sidx[t])) {
        svals[t] = v2;
        sidx[t] = i2;
      }
    }
    __syncthreads();
  }
  if (t == 0) out[b] = (float)sidx[0];
}

extern "C" void kernel_launch(void* const* d_in, const int* in_sizes, int n_in,
                              void* d_out, int out_size, void* d_ws,
                              size_t ws_size, hipStream_t stream) {
  (void)in_sizes; (void)n_in; (void)out_size; (void)ws_size;
  const float* x = (const float*)d_in[0];
  // d_in[1] = mask_size (scalar 6), baked in as KS.
  float* logits = (float*)d_ws;              // 64000 floats = 256 KB scratch
  float* out = (float*)d_out;                // 64 argmax indices as float

  pgp_channel_kernel<<<NBC / WAVES_PER_BLOCK, 256, 0, stream>>>(x, logits);
  pgp_argmax_kernel<<<NB, 256, 0, stream>>>(logits, out);
}